// TimeVaryingLPBiquad_62637803045374
// MI455X (gfx1250) — compile-verified
//
#include <hip/hip_runtime.h>
#include <cstdint>

#ifndef __has_builtin
#define __has_builtin(x) 0
#endif

#if __has_builtin(__builtin_amdgcn_global_load_async_to_lds_b128)
#define HAVE_ASYNC_LOAD 1
#else
#define HAVE_ASYNC_LOAD 0
#endif
#if __has_builtin(__builtin_amdgcn_global_store_async_from_lds_b128)
#define HAVE_ASYNC_STORE 1
#else
#define HAVE_ASYNC_STORE 0
#endif

#define THREADS 256
#define BB 256
#define TT 16384
#define TILE 4096
#define NTILE (TT / TILE)        // 4 tiles per row, state carried across tiles
#define CHUNK (TILE / THREADS)   // 16 samples per thread per tile
#define SP1 (TILE + (TILE / 64) * 4)  // +4 floats pad per 64 (keeps 16B alignment)
#define SP2 (TILE + (TILE / 64))      // float2 array pad

// log-domain modulation constants (fp32 literals of the reference doubles)
#define LOG_MIN_W (-4.6051701859880914f)   // ln(0.01)
#define LOG_MAX_W ( 1.1447290411865582f)   // ln(3.14159)
#define LOG_MIN_Q (-0.3465831812089371f)   // ln(0.7071)
#define LOG_MAX_Q ( 1.3862943611198906f)   // ln(4.0)
#define DLW (LOG_MAX_W - LOG_MIN_W)
#define DLQ (LOG_MAX_Q - LOG_MIN_Q)
#define STAB 0.999f                        // 1 - EPS
#define INV_STAB (1.0f / 0.999f)

__device__ __forceinline__ int padf(int i) { return i + ((i >> 6) << 2); }
__device__ __forceinline__ int pad2(int i) { return i + (i >> 6); }

// Types matching the builtin prototype: v4i with AS1 (global) / AS3 (LDS)
typedef int v4i __attribute__((vector_size(16)));
typedef __attribute__((address_space(1))) v4i glb_v4i;
typedef __attribute__((address_space(3))) v4i lds_v4i;

__device__ __forceinline__ lds_v4i* lds_v4i_addr(const void* p) {
  // generic LDS pointer low 32 bits == LDS byte address
  return (lds_v4i*)(uint32_t)(uint64_t)p;
}
__device__ __forceinline__ glb_v4i* glb_v4i_addr(const void* p) {
  return (glb_v4i*)(uint64_t)p;
}

// 16B global -> LDS, async on CDNA5 (tracked by ASYNCcnt)
__device__ __forceinline__ void copy16_g2l(const float* g, float* l) {
#if HAVE_ASYNC_LOAD
  __builtin_amdgcn_global_load_async_to_lds_b128(glb_v4i_addr(g), lds_v4i_addr(l), 0, 0);
#else
  *(float4*)l = *(const float4*)g;
#endif
}
// 16B LDS -> global, async on CDNA5
__device__ __forceinline__ void copy16_l2g(float* g, const float* l) {
#if HAVE_ASYNC_STORE
  __builtin_amdgcn_global_store_async_from_lds_b128(glb_v4i_addr(g), lds_v4i_addr(l), 0, 0);
#else
  *(float4*)g = *(const float4*)l;
#endif
}
__device__ __forceinline__ void wait_async_all() {
#if HAVE_ASYNC_LOAD || HAVE_ASYNC_STORE
#if __has_builtin(__builtin_amdgcn_s_wait_asynccnt)
  __builtin_amdgcn_s_wait_asynccnt(0);
#else
  asm volatile("s_wait_asynccnt 0" ::: "memory");
#endif
#endif
}

__global__ __launch_bounds__(THREADS)
void TimeVaryingLPBiquad_kernel(const float* __restrict__ x,
                                const float* __restrict__ wsig,
                                const float* __restrict__ qsig,
                                float* __restrict__ out_yab,
                                float* __restrict__ out_a,
                                float* __restrict__ out_b,
                                float* __restrict__ out_ya) {
  __shared__ float  s_x[2][SP1];          // x tile (double buffered), overwritten with y_a
  __shared__ float  s_yab[SP1];           // FIR output staging
  __shared__ float2 s_a12[SP2];           // (a1, a2) per sample
  __shared__ float  s_scan[THREADS][6];   // affine map scan: m00 m01 m10 m11 v0 v1

  const int row = blockIdx.x;
  const int tid = threadIdx.x;
  const float* xg = x    + (size_t)row * TT;
  const float* wg = wsig + (size_t)row * TT;
  const float* qg = qsig + (size_t)row * TT;
  float* yab_g = out_yab + (size_t)row * TT;
  float* ya_g  = out_ya  + (size_t)row * TT;
  float* ag    = out_a   + (size_t)row * TT * 3;
  float* bg    = out_b   + (size_t)row * TT * 3;

  float st0 = 0.0f, st1 = 0.0f;  // tile-entry state (y[t-1], y[t-2]); zero init per reference

  for (int tk = 0; tk < NTILE; ++tk) {
    const int tb = tk * TILE;
    float* xb = s_x[tk & 1];

    // (1) kick off async staging of this tile's x into LDS (overlaps phase 2)
    #pragma unroll
    for (int g2 = 0; g2 < TILE / (THREADS * 4); ++g2) {
      const int m = g2 * THREADS * 4 + tid * 4;
      copy16_g2l(xg + tb + m, &xb[padf(m)]);
    }

    // (2) coefficient generation (transcendental-heavy, fully parallel, coalesced)
    #pragma unroll
    for (int g2 = 0; g2 < TILE / (THREADS * 4); ++g2) {
      const int m  = g2 * THREADS * 4 + tid * 4;
      const int gi = tb + m;
      const float4 ws = *(const float4*)(wg + gi);
      const float4 qs = *(const float4*)(qg + gi);
      const float wv[4] = {ws.x, ws.y, ws.z, ws.w};
      const float qv[4] = {qs.x, qs.y, qs.z, qs.w};
      float a1v[4], a2v[4], b0v[4];
      #pragma unroll
      for (int c = 0; c < 4; ++c) {
        const float w  = __expf(fmaf(DLW, wv[c], LOG_MIN_W));
        const float rq = __expf(-fmaf(DLQ, qv[c], LOG_MIN_Q));  // 1/q directly
        const float sw = __sinf(w);
        const float cw = __cosf(w);
        const float alpha = 0.5f * sw * rq;
        const float ia0 = __builtin_amdgcn_rcpf(1.0f + alpha);
        const float a1 = -2.0f * cw * STAB * ia0;
        const float a2 = (1.0f - alpha) * STAB * ia0;
        const float b0 = 0.5f * (1.0f - cw) * ia0;
        a1v[c] = a1; a2v[c] = a2; b0v[c] = b0;
        s_a12[pad2(m + c)] = make_float2(a1, a2);
      }
      // a_full = {1, a1, a2} x4 -> 3 contiguous b128 stores
      *(float4*)(ag + (size_t)gi * 3 + 0) = make_float4(1.0f, a1v[0], a2v[0], 1.0f);
      *(float4*)(ag + (size_t)gi * 3 + 4) = make_float4(a1v[1], a2v[1], 1.0f, a1v[2]);
      *(float4*)(ag + (size_t)gi * 3 + 8) = make_float4(a2v[2], 1.0f, a1v[3], a2v[3]);
      // b = {b0, 2*b0, b0} x4
      *(float4*)(bg + (size_t)gi * 3 + 0) = make_float4(b0v[0], 2.0f * b0v[0], b0v[0], b0v[1]);
      *(float4*)(bg + (size_t)gi * 3 + 4) = make_float4(2.0f * b0v[1], b0v[1], b0v[2], 2.0f * b0v[2]);
      *(float4*)(bg + (size_t)gi * 3 + 8) = make_float4(b0v[2], b0v[3], 2.0f * b0v[3], b0v[3]);
    }

    // (3) wait own async ops (x loads done; prior tile's stores drained), then block sync
    wait_async_all();
    __syncthreads();

    // (4) per-thread chunk: compose 2x2 companion affine maps  s -> A_t s + (x_t,0)
    const int c0 = tid * CHUNK;
    float m00 = 1.f, m01 = 0.f, m10 = 0.f, m11 = 1.f, v0 = 0.f, v1 = 0.f;
    #pragma unroll
    for (int j = 0; j < CHUNK; ++j) {
      const int i = c0 + j;
      const float2 a = s_a12[pad2(i)];
      const float xv = xb[padf(i)];
      const float n0 = -fmaf(a.x, m00, a.y * m10);
      const float n1 = -fmaf(a.x, m01, a.y * m11);
      const float nv = xv - fmaf(a.x, v0, a.y * v1);
      m10 = m00; m11 = m01; v1 = v0;
      m00 = n0;  m01 = n1;  v0 = nv;
    }
    s_scan[tid][0] = m00; s_scan[tid][1] = m01; s_scan[tid][2] = m10;
    s_scan[tid][3] = m11; s_scan[tid][4] = v0;  s_scan[tid][5] = v1;
    __syncthreads();

    // (5) Hillis-Steele inclusive scan of affine maps across 256 threads (8 rounds)
    #pragma unroll
    for (int d = 1; d < THREADS; d <<= 1) {
      float p0, p1, p2, p3, p4, p5;
      const bool act = (tid >= d);
      if (act) {
        p0 = s_scan[tid - d][0]; p1 = s_scan[tid - d][1]; p2 = s_scan[tid - d][2];
        p3 = s_scan[tid - d][3]; p4 = s_scan[tid - d][4]; p5 = s_scan[tid - d][5];
      }
      __syncthreads();
      if (act) {  // new = cur ∘ prev : M = Mc*Mp ; v = Mc*vp + vc
        const float q00 = fmaf(m00, p0, m01 * p2);
        const float q01 = fmaf(m00, p1, m01 * p3);
        const float q10 = fmaf(m10, p0, m11 * p2);
        const float q11 = fmaf(m10, p1, m11 * p3);
        const float qv0 = fmaf(m00, p4, m01 * p5) + v0;
        const float qv1 = fmaf(m10, p4, m11 * p5) + v1;
        m00 = q00; m01 = q01; m10 = q10; m11 = q11; v0 = qv0; v1 = qv1;
        s_scan[tid][0] = m00; s_scan[tid][1] = m01; s_scan[tid][2] = m10;
        s_scan[tid][3] = m11; s_scan[tid][4] = v0;  s_scan[tid][5] = v1;
      }
      __syncthreads();
    }

    // incoming chunk state = agg[tid-1] applied to tile-entry state; also tile-exit state
    float y1, y2;
    if (tid == 0) { y1 = st0; y2 = st1; }
    else {
      y1 = fmaf(s_scan[tid - 1][0], st0, fmaf(s_scan[tid - 1][1], st1, s_scan[tid - 1][4]));
      y2 = fmaf(s_scan[tid - 1][2], st0, fmaf(s_scan[tid - 1][3], st1, s_scan[tid - 1][5]));
    }
    const float e0 = fmaf(s_scan[THREADS - 1][0], st0, fmaf(s_scan[THREADS - 1][1], st1, s_scan[THREADS - 1][4]));
    const float e1 = fmaf(s_scan[THREADS - 1][2], st0, fmaf(s_scan[THREADS - 1][3], st1, s_scan[THREADS - 1][5]));
    st0 = e0; st1 = e1;

    // (6) replay: IIR output + fused 3-tap FIR; b0 reconstructed from (a1,a2)
    #pragma unroll
    for (int j = 0; j < CHUNK; ++j) {
      const int i = c0 + j;
      const float2 a = s_a12[pad2(i)];
      const float xv = xb[padf(i)];
      const float y  = xv - fmaf(a.x, y1, a.y * y2);
      const float b0 = 0.25f * fmaf(a.x + a.y, INV_STAB, 1.0f);  // == (1-cosw)/(2*a0)
      const float yab = fmaf(b0, y + y2, (2.0f * b0) * y1);
      xb[padf(i)]    = y;     // overwrite x slot with y_a
      s_yab[padf(i)] = yab;
      y2 = y1; y1 = y;
    }
    __syncthreads();

    // (7) coalesced async drain of y_a and y_ab from LDS to global
    #pragma unroll
    for (int g2 = 0; g2 < TILE / (THREADS * 4); ++g2) {
      const int m = g2 * THREADS * 4 + tid * 4;
      copy16_l2g(ya_g + tb + m, &xb[padf(m)]);
      copy16_l2g(yab_g + tb + m, &s_yab[padf(m)]);
    }
  }
  wait_async_all();
}

extern "C" void kernel_launch(void* const* d_in, const int* in_sizes, int n_in,
                              void* d_out, int out_size, void* d_ws, size_t ws_size,
                              hipStream_t stream) {
  (void)in_sizes; (void)n_in; (void)out_size; (void)d_ws; (void)ws_size;
  const float* x = (const float*)d_in[0];
  const float* w = (const float*)d_in[1];
  const float* q = (const float*)d_in[2];
  float* o = (float*)d_out;
  float* yab   = o;                                    // [B,T]
  float* afull = o + (size_t)BB * TT;                  // [B,T,3]
  float* bco   = afull + (size_t)BB * TT * 3;          // [B,T,3]
  float* ya    = bco + (size_t)BB * TT * 3;            // [B,T]
  hipLaunchKernelGGL(TimeVaryingLPBiquad_kernel, dim3(BB), dim3(THREADS), 0, stream,
                     x, w, q, yab, afull, bco, ya);
}